// PA_Module_16836271800508
// MI455X (gfx1250) — compile-verified
//
#include <hip/hip_runtime.h>
#include <math.h>

// ---------------------------------------------------------------------------
// Self-attention (PA module), MI455X / gfx1250, wave32, fp32 WMMA path.
//
//   B=4, C=64, MID=8, N=H*W=4096
//   q = Wq x + bq   [B,8,N]     stored K-pair-interleaved: [B][4][N][2]
//   k = Wk x + bk   [B,8,N]     stored K-pair-interleaved: [B][4][N][2]
//   v = Wv x + bv   [B,64,N]    stored transposed+pair-interleaved:
//                               [B][N/2][64][2]  (pairs of rows m, m+1)
//   att = softmax_m(q^T k);  out = gamma * (att @ v^T) + x
//
// Flash-attention structure: never materialize the 4096x4096 att matrix
// (256 MB of HBM waste at 23.3 TB/s); stream 64-wide K/V tiles from L2,
// online softmax in registers, V_WMMA_F32_16X16X4_F32 for every matmul.
// The pair-interleaved layouts make every WMMA B-operand (and the Q
// A-operand) a single aligned global_load_b64 -> 1 load per WMMA in the
// hot P*V^T loop instead of 2.
// ---------------------------------------------------------------------------

#define BB   4
#define CC   64
#define MIDC 8
#define NN   4096

typedef float v2f __attribute__((ext_vector_type(2)));
typedef float v8f __attribute__((ext_vector_type(8)));

static __device__ __forceinline__ v8f wmma_f32_k4(v2f a, v2f b, v8f c) {
  // D(16x16,f32) = A(16x4,f32) * B(4x16,f32) + C
  return __builtin_amdgcn_wmma_f32_16x16x4_f32(
      /*neg_a=*/false, a, /*neg_b=*/false, b,
      /*c_mod=*/(short)0, c, /*reuse_a=*/false, /*reuse_b=*/false);
}

// Padded LDS row stride (floats): 68 = even (paired K-operand loads merge to
// ds_load_b64 / 2addr_b64) and 68*l mod 64 banks = 4*l -> 16 lanes of a half
// hit 16 distinct banks; the two halves differ by 544 mod 64 = 32 banks.
#define WPAD 68

// ---------------------------------------------------------------------------
// Kernel 1: QKV projections as WMMA GEMMs. One wave per 16-column n-tile,
// 8 waves/block sharing one LDS copy of the weights.
//   D_qk[n][c'] (c'=0..7 -> q, 8..15 -> k) = sum_c x^T[n][c] * Wqk[c'][c]
//   D_v [n][o]  (4 o-chunks)               = sum_c x^T[n][c] * Wv[o][c]
// K=64 contracted as 16 chunks of K=4.  5 WMMAs per chunk per wave.
// ---------------------------------------------------------------------------
__global__ __launch_bounds__(256) void qkv_proj_kernel(
    const float* __restrict__ x,
    const float* __restrict__ Wq, const float* __restrict__ bq,
    const float* __restrict__ Wk, const float* __restrict__ bk,
    const float* __restrict__ Wv, const float* __restrict__ bv,
    float* __restrict__ qP, float* __restrict__ kP, float* __restrict__ vP) {
  __shared__ float sWqk[16 * WPAD];   // rows 0-7 = Wq, rows 8-15 = Wk
  __shared__ float sWv[CC * WPAD];
  __shared__ float sb[2 * MIDC + CC]; // bq | bk | bv

  const int tid = threadIdx.x;
  for (int i = tid; i < 16 * CC; i += 256) {
    const int r = i >> 6, c = i & 63;
    sWqk[r * WPAD + c] = (r < MIDC) ? Wq[r * CC + c] : Wk[(r - MIDC) * CC + c];
  }
  for (int i = tid; i < CC * CC; i += 256)
    sWv[(i >> 6) * WPAD + (i & 63)] = Wv[i];
  if (tid < 2 * MIDC + CC) {
    sb[tid] = (tid < MIDC) ? bq[tid]
            : (tid < 2 * MIDC) ? bk[tid - MIDC]
                               : bv[tid - 2 * MIDC];
  }
  __syncthreads();

  const int wave = tid >> 5;
  const int lane = tid & 31;
  const int h = lane >> 4;          // WMMA K-half / row-half selector
  const int l = lane & 15;

  const int tile = blockIdx.x * 8 + wave;   // 0 .. B*N/16-1 (no batch split)
  const int b = tile >> 8;
  const int n0 = (tile & 255) << 4;

  const float* xb = x + (size_t)b * CC * NN;

  v8f dqk = {};
  v8f dv[4] = {};
#pragma unroll
  for (int kc = 0; kc < 16; ++kc) {
    const int c0 = 4 * kc;
    // A operand: x^T tile, lane = row n, VGPR0/1 = K = 2h / 2h+1.
    v2f ax;
    ax.x = xb[(size_t)(c0 + 2 * h) * NN + n0 + l];
    ax.y = xb[(size_t)(c0 + 2 * h + 1) * NN + n0 + l];
    // B operands from LDS (lane = output column), paired loads -> b64.
    const v2f bqk = *(const v2f*)&sWqk[l * WPAD + c0 + 2 * h];
    dqk = wmma_f32_k4(ax, bqk, dqk);
#pragma unroll
    for (int t = 0; t < 4; ++t) {
      const v2f bw = *(const v2f*)&sWv[(16 * t + l) * WPAD + c0 + 2 * h];
      dv[t] = wmma_f32_k4(ax, bw, dv[t]);
    }
  }

  // Stores. D layout: VGPR r -> rows n0+r (h=0) / n0+r+8 (h=1), lane%16 = col.
  // q/k pair-interleaved: [B][4][N][2], channel c' -> (pair c'>>1, elem c'&1).
  {
    const float biasqk = sb[l];
    const int ch = (l < MIDC) ? l : (l - MIDC);
    float* base = (l < MIDC) ? qP : kP;
    float* dst = base + (((size_t)b * (MIDC / 2) + (ch >> 1)) * NN) * 2 + (ch & 1);
#pragma unroll
    for (int r = 0; r < 8; ++r) {
      const int n = n0 + r + 8 * h;
      dst[(size_t)n * 2] = dqk[r] + biasqk;
    }
  }
  // v pair-interleaved: [B][N/2][C][2]; rows n0+r+8h and +1 pair in-register.
  float* vb = vP + ((size_t)b * (NN / 2)) * CC * 2;
#pragma unroll
  for (int t = 0; t < 4; ++t) {
    const float biasv = sb[2 * MIDC + 16 * t + l];
#pragma unroll
    for (int r = 0; r < 8; r += 2) {
      const int n = n0 + r + 8 * h;             // even
      v2f pair;
      pair.x = dv[t][r] + biasv;
      pair.y = dv[t][r + 1] + biasv;
      *(v2f*)&vb[((size_t)(n >> 1) * CC + 16 * t + l) * 2] = pair;  // b64 store
    }
  }
}

// ---------------------------------------------------------------------------
// Kernel 2: flash attention. One wave per (batch, 16-row query tile):
// 4*256 = 1024 waves. Per 64-wide key tile: 8 WMMAs for the four S sub-tiles
// (B operands: one b64 load each), elementwise 4->1 tile reduction then ONE
// butterfly per row (masks 1,2,4,8 stay inside each 16-lane half = the C/D
// row grouping), online-softmax rescale, LDS bounce to re-layout P from C/D
// layout (lane=col) into A layout (lane=row), then 64 WMMAs for O += P*V^T
// with exactly one b64 V load per WMMA. EXEC all-ones at every WMMA.
// ---------------------------------------------------------------------------
__global__ __launch_bounds__(32) void flash_attn_kernel(
    const float* __restrict__ qP, const float* __restrict__ kP,
    const float* __restrict__ vP, const float* __restrict__ x,
    const float* __restrict__ gamma, float* __restrict__ out) {
  __shared__ float Plds[16 * WPAD];   // 16 rows x 64 cols of P, padded

  const int wg = blockIdx.x;
  const int b = wg >> 8;            // / (NN/16)
  const int n0 = (wg & 255) << 4;   // query tile base
  const int lane = threadIdx.x;
  const int h = lane >> 4;
  const int l = lane & 15;

  const float* qb = qP + (size_t)b * MIDC * NN;       // [4][N][2]
  const float* kb = kP + (size_t)b * MIDC * NN;       // [4][N][2]
  const float* vb = vP + (size_t)b * (NN / 2) * CC * 2;  // [N/2][C][2]

  // Q tile as WMMA A operand: channels (2h, 2h+1) = interleaved pair -> b64.
  const v2f aq0 = *(const v2f*)&qb[((size_t)h * NN + n0 + l) * 2];
  const v2f aq1 = *(const v2f*)&qb[((size_t)(2 + h) * NN + n0 + l) * 2];

  v8f o_acc[4] = {};                 // rows n (VGPR r -> n = n0+r+8h),
                                     // cols c (lane = c%16), 4 chunks of 16
  float m_run[8], l_run[8];
#pragma unroll
  for (int r = 0; r < 8; ++r) { m_run[r] = -INFINITY; l_run[r] = 0.0f; }

  for (int m0 = 0; m0 < NN; m0 += 64) {
    if (m0 + 64 < NN) {  // prefetch next K/V tile toward the WGP
      __builtin_prefetch(&kb[((size_t)h * NN + m0 + 64 + l) * 2], 0, 3);
      __builtin_prefetch(&vb[(size_t)(m0 / 2 + 32 + lane) * CC * 2], 0, 3);
    }

    // ---- S = Q^T K for four 16-wide sub-tiles --------------------------
    v8f s[4];
#pragma unroll
    for (int u = 0; u < 4; ++u) {
      const int m = m0 + 16 * u;
      // B(4x16): VGPR = K%2, half = K/2, lane%16 = col m. Channel pairs
      // (2h,2h+1) and (4+2h,5+2h) are interleaved in kP -> single b64 loads.
      const v2f bk0 = *(const v2f*)&kb[((size_t)h * NN + m + l) * 2];
      const v2f bk1 = *(const v2f*)&kb[((size_t)(2 + h) * NN + m + l) * 2];
      v8f acc = {};
      acc = wmma_f32_k4(aq0, bk0, acc);
      s[u] = wmma_f32_k4(aq1, bk1, acc);
    }

    // ---- online softmax over the 64 columns ----------------------------
    float scale[8];
#pragma unroll
    for (int r = 0; r < 8; ++r) {
      // 4 tiles -> 1 in VALU, then one butterfly across the 16-lane half.
      float mx = fmaxf(fmaxf(s[0][r], s[1][r]), fmaxf(s[2][r], s[3][r]));
      mx = fmaxf(mx, __shfl_xor(mx, 1, 32));
      mx = fmaxf(mx, __shfl_xor(mx, 2, 32));
      mx = fmaxf(mx, __shfl_xor(mx, 4, 32));
      mx = fmaxf(mx, __shfl_xor(mx, 8, 32));
      const float mn = fmaxf(m_run[r], mx);
      scale[r] = __expf(m_run[r] - mn);
      m_run[r] = mn;
      float rsum = 0.0f;
#pragma unroll
      for (int u = 0; u < 4; ++u) {
        const float pe = __expf(s[u][r] - mn);
        Plds[(r + 8 * h) * WPAD + 16 * u + l] = pe;  // P in [row][col] layout
        rsum += pe;
      }
      rsum += __shfl_xor(rsum, 1, 32);
      rsum += __shfl_xor(rsum, 2, 32);
      rsum += __shfl_xor(rsum, 4, 32);
      rsum += __shfl_xor(rsum, 8, 32);
      l_run[r] = l_run[r] * scale[r] + rsum;
    }

    // Rescale running output (row mapping matches scale[] halves).
#pragma unroll
    for (int t = 0; t < 4; ++t)
#pragma unroll
      for (int r = 0; r < 8; ++r) o_acc[t][r] *= scale[r];

    __syncthreads();  // single-wave WG: degrades to S_NOP; DS is in-order

    // ---- O += P * V^T: 16 K-chunks of 4 over the 64 columns ------------
#pragma unroll
    for (int jj = 0; jj < 16; ++jj) {
      // A operand: P re-laid out lane=row; paired K values -> ds b64 load.
      const v2f ap = *(const v2f*)&Plds[l * WPAD + 4 * jj + 2 * h];
      // B operand: rows (m, m+1), m = m0+4jj+2h even -> interleaved pair in
      // vP at pair-index m/2 -> single b64 load, contiguous across the half.
      const size_t mp = (size_t)(m0 / 2 + 2 * jj + h);
#pragma unroll
      for (int t = 0; t < 4; ++t) {
        const v2f bv2 = *(const v2f*)&vb[(mp * CC + 16 * t + l) * 2];
        o_acc[t] = wmma_f32_k4(ap, bv2, o_acc[t]);
      }
    }
    __syncthreads();
  }

  // Epilogue: normalize by softmax denominator, gamma-scale, residual add.
  const float g = gamma[0];
#pragma unroll
  for (int t = 0; t < 4; ++t) {
#pragma unroll
    for (int r = 0; r < 8; ++r) {
      const int n = n0 + r + 8 * h;
      const int c = 16 * t + l;
      const size_t idx = ((size_t)b * CC + c) * NN + n;
      out[idx] = g * (o_acc[t][r] / l_run[r]) + x[idx];
    }
  }
}

// ---------------------------------------------------------------------------
// Host launcher. Workspace: qP (512 KB) | kP (512 KB) | vP (4 MB) = 5 MB.
// ---------------------------------------------------------------------------
extern "C" void kernel_launch(void* const* d_in, const int* in_sizes, int n_in,
                              void* d_out, int out_size, void* d_ws,
                              size_t ws_size, hipStream_t stream) {
  (void)in_sizes; (void)n_in; (void)out_size; (void)ws_size;
  const float* x     = (const float*)d_in[0];
  const float* Wq    = (const float*)d_in[1];
  const float* bq    = (const float*)d_in[2];
  const float* Wk    = (const float*)d_in[3];
  const float* bk    = (const float*)d_in[4];
  const float* Wv    = (const float*)d_in[5];
  const float* bv    = (const float*)d_in[6];
  const float* gamma = (const float*)d_in[7];
  float* out = (float*)d_out;

  float* qws = (float*)d_ws;                       // [B][4][N][2]
  float* kws = qws + (size_t)BB * MIDC * NN;       // [B][4][N][2]
  float* vws = kws + (size_t)BB * MIDC * NN;       // [B][N/2][C][2]

  qkv_proj_kernel<<<(BB * (NN / 16)) / 8, 256, 0, stream>>>(
      x, Wq, bq, Wk, bk, Wv, bv, qws, kws, vws);

  flash_attn_kernel<<<BB * (NN / 16), 32, 0, stream>>>(
      qws, kws, vws, x, gamma, out);
}